// PointNet2SSG_45586782880368
// MI455X (gfx1250) — compile-verified
//
#include <hip/hip_runtime.h>
#include <hip/hip_bf16.h>

typedef float v2f __attribute__((ext_vector_type(2)));
typedef float v8f __attribute__((ext_vector_type(8)));

// ---------------------------------------------------------------------------
// Generic fused-MLP layer on WMMA f32 16x16x4, using pre-padded weights.
// X:  LDS activations (rows x cinp, zero padded columns >= true cin)
// Wp: global weights pre-padded to (cout x cinp), zeros in pad columns
// H:  LDS output (rows x cout) = relu((X*Wp^T) * scale + shift)
// A-frag (ISA 7.12.2): lane l holds row (l%16), K = kb + 2*(l/16)+{0,1}
// B-frag mirrors with column (l%16). C/D: vgpr r, lane l -> (M=r+8*(l/16), N=l%16)
// All fragment loads are aligned 8-byte vector loads (cinp % 4 == 0, k even).
// ---------------------------------------------------------------------------
__device__ __forceinline__ void mlp_layer(
    const float* __restrict__ X, int ldx, int cinp,
    const float* __restrict__ Wp,
    const float* __restrict__ scale, const float* __restrict__ shift,
    float* __restrict__ H, int ldh,
    int rows, int cout, int wave, int nwaves, int lane)
{
    const int half = lane >> 4;
    const int lm   = lane & 15;
    const int mtiles = rows >> 4, ntiles = cout >> 4;
    for (int t = wave; t < mtiles * ntiles; t += nwaves) {
        const int mt = t / ntiles, nt = t - mt * ntiles;
        const int n    = nt * 16 + lm;
        const int arow = mt * 16 + lm;
        const float* xr = X  + arow * ldx  + 2 * half;
        const float* wr = Wp + n    * cinp + 2 * half;
        v8f acc = {};
#pragma unroll 4
        for (int kb = 0; kb < cinp; kb += 4) {
            v2f a = *(const v2f*)(xr + kb);
            v2f b = *(const v2f*)(wr + kb);
            acc = __builtin_amdgcn_wmma_f32_16x16x4_f32(
                false, a, false, b, (short)0, acc, false, false);
        }
        const float sc = scale[n], sh = shift[n];
        const int mrow = mt * 16 + half * 8;
#pragma unroll
        for (int r = 0; r < 8; ++r) {
            float v = fmaxf(acc[r] * sc + sh, 0.0f);
            H[(mrow + r) * ldh + n] = v;
        }
    }
}

// Final layer fused with max over rows; outmax is a u32 bitwise-max buffer
// (valid float max for non-negative post-ReLU values; must be pre-zeroed).
__device__ __forceinline__ void mlp_layer_max(
    const float* __restrict__ X, int ldx, int cinp,
    const float* __restrict__ Wp,
    const float* __restrict__ scale, const float* __restrict__ shift,
    unsigned int* outmax,
    int rows, int cout, int wave, int nwaves, int lane)
{
    const int half = lane >> 4;
    const int lm   = lane & 15;
    const int mtiles = rows >> 4, ntiles = cout >> 4;
    for (int t = wave; t < mtiles * ntiles; t += nwaves) {
        const int mt = t / ntiles, nt = t - mt * ntiles;
        const int n    = nt * 16 + lm;
        const int arow = mt * 16 + lm;
        const float* xr = X  + arow * ldx  + 2 * half;
        const float* wr = Wp + n    * cinp + 2 * half;
        v8f acc = {};
#pragma unroll 4
        for (int kb = 0; kb < cinp; kb += 4) {
            v2f a = *(const v2f*)(xr + kb);
            v2f b = *(const v2f*)(wr + kb);
            acc = __builtin_amdgcn_wmma_f32_16x16x4_f32(
                false, a, false, b, (short)0, acc, false, false);
        }
        const float sc = scale[n], sh = shift[n];
        float m = 0.0f;
#pragma unroll
        for (int r = 0; r < 8; ++r)
            m = fmaxf(m, fmaxf(acc[r] * sc + sh, 0.0f));
        m = fmaxf(m, __shfl_xor(m, 16));      // combine the two row halves
        if (half == 0)
            atomicMax(outmax + n, __float_as_uint(m));
    }
}

// ---------------------------------------------------------------------------
// Pad weights (cout x cin) -> (cout x cinp) with zero columns.
// ---------------------------------------------------------------------------
__global__ void pad_w_kernel(const float* __restrict__ W, int cin, int cinp,
                             int total, float* __restrict__ Wp)
{
    int i = blockIdx.x * blockDim.x + threadIdx.x;
    if (i < total) {
        int n = i / cinp, k = i - n * cinp;
        Wp[i] = (k < cin) ? W[n * cin + k] : 0.0f;
    }
}

// ---------------------------------------------------------------------------
// Fold BN: scale = gamma*rsqrt(var+eps); shift = (bias-mean)*scale + beta
// ---------------------------------------------------------------------------
__global__ void fold_bn_kernel(const float* __restrict__ bias,
                               const float* __restrict__ gamma,
                               const float* __restrict__ beta,
                               const float* __restrict__ mean,
                               const float* __restrict__ var,
                               int cout, float* __restrict__ scale,
                               float* __restrict__ shift)
{
    int i = blockIdx.x * blockDim.x + threadIdx.x;
    if (i < cout) {
        float s = gamma[i] * rsqrtf(var[i] + 1e-5f);
        scale[i] = s;
        shift[i] = (bias[i] - mean[i]) * s + beta[i];
    }
}

__global__ void zero_kernel(unsigned int* __restrict__ p, size_t n)
{
    size_t i = (size_t)blockIdx.x * blockDim.x + threadIdx.x;
    if (i < n) p[i] = 0u;
}

// ---------------------------------------------------------------------------
// Farthest point sampling (one block per batch) + gather of new_xyz.
// Matches jnp.argmax first-occurrence tie-break (smallest index wins ties).
// ---------------------------------------------------------------------------
template <int N, int NPOINT, int THREADS>
__global__ __launch_bounds__(THREADS) void fps_kernel(
    const float* __restrict__ xyz, int* __restrict__ fidx,
    float* __restrict__ nxyz)
{
    constexpr int PPT = N / THREADS;
    __shared__ float sv[THREADS];
    __shared__ int   si[THREADS];
    __shared__ float scent[3];
    const int b = blockIdx.x, tid = threadIdx.x;
    const float* base = xyz + (size_t)b * N * 3;

    float px[PPT], py[PPT], pz[PPT], dist[PPT];
#pragma unroll
    for (int i = 0; i < PPT; ++i) {
        int j = tid + i * THREADS;
        px[i] = base[j * 3 + 0];
        py[i] = base[j * 3 + 1];
        pz[i] = base[j * 3 + 2];
        dist[i] = 1e10f;
    }
    int far = 0;
    for (int it = 0; it < NPOINT; ++it) {
        if (tid == 0) {
            fidx[(size_t)b * NPOINT + it] = far;
            scent[0] = base[far * 3 + 0];
            scent[1] = base[far * 3 + 1];
            scent[2] = base[far * 3 + 2];
        }
        __syncthreads();
        const float cx = scent[0], cy = scent[1], cz = scent[2];
        float bestv = -1.0f; int besti = 0;
#pragma unroll
        for (int i = 0; i < PPT; ++i) {
            int j = tid + i * THREADS;
            float dx = px[i] - cx, dy = py[i] - cy, dz = pz[i] - cz;
            float d = dx * dx + dy * dy + dz * dz;
            float nd = fminf(dist[i], d);
            dist[i] = nd;
            if (nd > bestv) { bestv = nd; besti = j; }  // ascending j keeps first max
        }
        sv[tid] = bestv; si[tid] = besti;
        __syncthreads();
        for (int off = THREADS / 2; off > 0; off >>= 1) {
            if (tid < off) {
                float v2 = sv[tid + off]; int i2 = si[tid + off];
                if (v2 > sv[tid] || (v2 == sv[tid] && i2 < si[tid])) {
                    sv[tid] = v2; si[tid] = i2;
                }
            }
            __syncthreads();
        }
        far = si[0];
        __syncthreads();
    }
    for (int p = tid; p < NPOINT; p += THREADS) {
        int j = fidx[(size_t)b * NPOINT + p];
        nxyz[((size_t)b * NPOINT + p) * 3 + 0] = base[j * 3 + 0];
        nxyz[((size_t)b * NPOINT + p) * 3 + 1] = base[j * 3 + 1];
        nxyz[((size_t)b * NPOINT + p) * 3 + 2] = base[j * 3 + 2];
    }
}

// ---------------------------------------------------------------------------
// Ball query: one wave per center. Collect first 64 valid indices ascending,
// pad with the first valid index (reference semantics).
// ---------------------------------------------------------------------------
__global__ __launch_bounds__(128) void ballq_kernel(
    const float* __restrict__ xyz, const float* __restrict__ nxyz,
    int* __restrict__ idx, int N, int S, int total, float r2)
{
    const int gw = blockIdx.x * (blockDim.x >> 5) + (threadIdx.x >> 5);
    if (gw >= total) return;
    const int lane = threadIdx.x & 31;
    const int b = gw / S;
    const float* base = xyz + (size_t)b * N * 3;
    const float cx = nxyz[(size_t)gw * 3 + 0];
    const float cy = nxyz[(size_t)gw * 3 + 1];
    const float cz = nxyz[(size_t)gw * 3 + 2];
    int* out = idx + (size_t)gw * 64;

    int cnt = 0, first = 0;
    for (int j0 = 0; j0 < N && cnt < 64; j0 += 32) {
        const int j = j0 + lane;
        bool valid = false;
        if (j < N) {
            float dx = base[j * 3 + 0] - cx;
            float dy = base[j * 3 + 1] - cy;
            float dz = base[j * 3 + 2] - cz;
            valid = (dx * dx + dy * dy + dz * dz) <= r2;
        }
        unsigned long long m = __ballot(valid);
        int pre  = __popcll(m & ((1ull << lane) - 1ull));
        int slot = cnt + pre;
        if (valid && slot < 64) out[slot] = j;
        if (cnt == 0 && m != 0ull) first = j0 + __ffsll((unsigned long long)m) - 1;
        cnt += __popcll(m);
    }
    if (cnt > 64) cnt = 64;
    for (int s = cnt + lane; s < 64; s += 32) out[s] = first;
}

// ---------------------------------------------------------------------------
// Module 1: one block per group (b, center). 64 samples, 3 -> 64 -> 64 -> 128.
// ---------------------------------------------------------------------------
__global__ __launch_bounds__(128) void sa1_mlp_kernel(
    const float* __restrict__ xyz, const int* __restrict__ idx1,
    const float* __restrict__ nxyz1,
    const float* W1, const float* sc1, const float* sh1,
    const float* W2, const float* sc2, const float* sh2,
    const float* W3, const float* sc3, const float* sh3,
    float* __restrict__ feats1)
{
    __shared__ float Xs[64 * 4];
    __shared__ float H1[64 * 64];
    __shared__ float H2[64 * 64];
    __shared__ unsigned int mb[128];
    const int blk = blockIdx.x;          // b*512 + c
    const int b = blk >> 9;
    const int tid = threadIdx.x, lane = tid & 31, wave = tid >> 5;

    if (tid < 64) {
        int j = idx1[(size_t)blk * 64 + tid];
        const float* p = xyz + ((size_t)b * 4096 + j) * 3;
        const float* c = nxyz1 + (size_t)blk * 3;
        Xs[tid * 4 + 0] = p[0] - c[0];
        Xs[tid * 4 + 1] = p[1] - c[1];
        Xs[tid * 4 + 2] = p[2] - c[2];
        Xs[tid * 4 + 3] = 0.0f;
    }
    mb[tid] = 0u;
    __syncthreads();
    mlp_layer(Xs, 4, 4,    W1, sc1, sh1, H1, 64, 64, 64, wave, 4, lane);
    __syncthreads();
    mlp_layer(H1, 64, 64,  W2, sc2, sh2, H2, 64, 64, 64, wave, 4, lane);
    __syncthreads();
    mlp_layer_max(H2, 64, 64, W3, sc3, sh3, mb, 64, 128, wave, 4, lane);
    __syncthreads();
    if (tid < 128)
        feats1[(size_t)blk * 128 + tid] = __uint_as_float(mb[tid]);
}

// ---------------------------------------------------------------------------
// Module 2: two blocks per group (32 rows each). 131 -> 128 -> 128 -> 256.
// LDS: A holds X(ld132) then H2(ld128); B holds H1.
// ---------------------------------------------------------------------------
__global__ __launch_bounds__(128) void sa2_mlp_kernel(
    const float* __restrict__ nxyz1, const float* __restrict__ feats1,
    const int* __restrict__ idx2, const float* __restrict__ nxyz2,
    const float* W1, const float* sc1, const float* sh1,
    const float* W2, const float* sc2, const float* sh2,
    const float* W3, const float* sc3, const float* sh3,
    unsigned int* __restrict__ feats2max)
{
    __shared__ float A[32 * 132];
    __shared__ float B[32 * 128];
    const int blk = blockIdx.x;
    const int sub = blk & 1;
    const int g = blk >> 1;              // b*128 + c
    const int b = g >> 7;
    const int tid = threadIdx.x, lane = tid & 31, wave = tid >> 5;

    for (int i = tid; i < 32 * 132; i += 128) {
        const int r = i / 132, col = i - r * 132;
        const int s = sub * 32 + r;
        const int j = idx2[(size_t)g * 64 + s];
        float v;
        if (col < 3)
            v = nxyz1[((size_t)b * 512 + j) * 3 + col] - nxyz2[(size_t)g * 3 + col];
        else if (col < 131)
            v = feats1[((size_t)b * 512 + j) * 128 + (col - 3)];
        else
            v = 0.0f;
        A[r * 132 + col] = v;
    }
    __syncthreads();
    mlp_layer(A, 132, 132, W1, sc1, sh1, B, 128, 32, 128, wave, 4, lane);
    __syncthreads();
    mlp_layer(B, 128, 128, W2, sc2, sh2, A, 128, 32, 128, wave, 4, lane);
    __syncthreads();
    mlp_layer_max(A, 128, 128, W3, sc3, sh3,
                  feats2max + (size_t)g * 256, 32, 256, wave, 4, lane);
}

// ---------------------------------------------------------------------------
// Module 3: 8 blocks per batch (16 rows each). 259 -> 256 -> 512 -> 1024.
// LDS: A holds X(ld260) then H2(ld512); B holds H1(ld256).
// ---------------------------------------------------------------------------
__global__ __launch_bounds__(128) void head_mlp_kernel(
    const float* __restrict__ nxyz2, const float* __restrict__ feats2,
    const float* W1, const float* sc1, const float* sh1,
    const float* W2, const float* sc2, const float* sh2,
    const float* W3, const float* sc3, const float* sh3,
    unsigned int* __restrict__ outmax)
{
    __shared__ float A[16 * 512];
    __shared__ float B[16 * 256];
    const int b = blockIdx.x >> 3;
    const int q = blockIdx.x & 7;
    const int tid = threadIdx.x, lane = tid & 31, wave = tid >> 5;

    for (int i = tid; i < 16 * 260; i += 128) {
        const int r = i / 260, col = i - r * 260;
        const int p = q * 16 + r;
        float v;
        if (col < 3)
            v = nxyz2[((size_t)b * 128 + p) * 3 + col];
        else if (col < 259)
            v = feats2[((size_t)b * 128 + p) * 256 + (col - 3)];
        else
            v = 0.0f;
        A[r * 260 + col] = v;
    }
    __syncthreads();
    mlp_layer(A, 260, 260, W1, sc1, sh1, B, 256, 16, 256, wave, 4, lane);
    __syncthreads();
    mlp_layer(B, 256, 256, W2, sc2, sh2, A, 512, 16, 512, wave, 4, lane);
    __syncthreads();
    mlp_layer_max(A, 512, 512, W3, sc3, sh3,
                  outmax + (size_t)b * 1024, 16, 1024, wave, 4, lane);
}

// ---------------------------------------------------------------------------
extern "C" void kernel_launch(void* const* d_in, const int* in_sizes, int n_in,
                              void* d_out, int out_size, void* d_ws, size_t ws_size,
                              hipStream_t stream)
{
    (void)in_sizes; (void)n_in; (void)out_size; (void)ws_size;
    const float* pc = (const float*)d_in[0];                 // (32,4096,3)
    // params flattened: per layer (w,b,gamma,beta,mean,var), 9 layers
    const float* W[9], *Bp[9], *G[9], *Be[9], *Mn[9], *Vr[9];
    for (int l = 0; l < 9; ++l) {
        W[l]  = (const float*)d_in[1 + l * 6 + 0];
        Bp[l] = (const float*)d_in[1 + l * 6 + 1];
        G[l]  = (const float*)d_in[1 + l * 6 + 2];
        Be[l] = (const float*)d_in[1 + l * 6 + 3];
        Mn[l] = (const float*)d_in[1 + l * 6 + 4];
        Vr[l] = (const float*)d_in[1 + l * 6 + 5];
    }
    static const int CIN [9] = {3,  64, 64,  131, 128, 128, 259, 256, 512};
    static const int CINP[9] = {4,  64, 64,  132, 128, 128, 260, 256, 512};
    static const int COUT[9] = {64, 64, 128, 128, 128, 256, 256, 512, 1024};

    // ---- workspace layout (floats) ----
    float* ws = (float*)d_ws;
    size_t off = 0;
    float* scale[9]; float* shift[9]; float* Wp[9];
    for (int l = 0; l < 9; ++l) { scale[l] = ws + off; off += COUT[l]; }
    for (int l = 0; l < 9; ++l) { shift[l] = ws + off; off += COUT[l]; }
    for (int l = 0; l < 9; ++l) { Wp[l] = ws + off; off += (size_t)COUT[l] * CINP[l]; }
    int*   fidx1  = (int*)(ws + off); off += (size_t)32 * 512;
    float* nxyz1  = ws + off;         off += (size_t)32 * 512 * 3;
    int*   idx1   = (int*)(ws + off); off += (size_t)32 * 512 * 64;
    float* feats1 = ws + off;         off += (size_t)32 * 512 * 128;
    int*   fidx2  = (int*)(ws + off); off += (size_t)32 * 128;
    float* nxyz2  = ws + off;         off += (size_t)32 * 128 * 3;
    int*   idx2   = (int*)(ws + off); off += (size_t)32 * 128 * 64;
    float* feats2 = ws + off;         off += (size_t)32 * 128 * 256;

    // ---- fold BN and pad weights ----
    for (int l = 0; l < 9; ++l) {
        int g = (COUT[l] + 255) / 256;
        fold_bn_kernel<<<g, 256, 0, stream>>>(Bp[l], G[l], Be[l], Mn[l], Vr[l],
                                              COUT[l], scale[l], shift[l]);
        int total = COUT[l] * CINP[l];
        pad_w_kernel<<<(total + 255) / 256, 256, 0, stream>>>(
            W[l], CIN[l], CINP[l], total, Wp[l]);
    }

    // ---- stage 1: FPS(4096 -> 512), ball query r=0.2, grouped MLP ----
    fps_kernel<4096, 512, 256><<<32, 256, 0, stream>>>(pc, fidx1, nxyz1);
    {
        int total = 32 * 512;                 // waves
        ballq_kernel<<<(total + 3) / 4, 128, 0, stream>>>(
            pc, nxyz1, idx1, 4096, 512, total, 0.2f * 0.2f);
    }
    sa1_mlp_kernel<<<32 * 512, 128, 0, stream>>>(
        pc, idx1, nxyz1,
        Wp[0], scale[0], shift[0], Wp[1], scale[1], shift[1],
        Wp[2], scale[2], shift[2], feats1);

    // ---- stage 2: FPS(512 -> 128), ball query r=0.4, grouped MLP ----
    fps_kernel<512, 128, 128><<<32, 128, 0, stream>>>(nxyz1, fidx2, nxyz2);
    {
        int total = 32 * 128;
        ballq_kernel<<<(total + 3) / 4, 128, 0, stream>>>(
            nxyz1, nxyz2, idx2, 512, 128, total, 0.4f * 0.4f);
    }
    {
        size_t n = (size_t)32 * 128 * 256;
        zero_kernel<<<(unsigned)((n + 255) / 256), 256, 0, stream>>>(
            (unsigned int*)feats2, n);
    }
    sa2_mlp_kernel<<<32 * 128 * 2, 128, 0, stream>>>(
        nxyz1, feats1, idx2, nxyz2,
        Wp[3], scale[3], shift[3], Wp[4], scale[4], shift[4],
        Wp[5], scale[5], shift[5], (unsigned int*)feats2);

    // ---- stage 3: head MLP + global max over 128 points ----
    {
        size_t n = (size_t)32 * 1024;
        zero_kernel<<<(unsigned)((n + 255) / 256), 256, 0, stream>>>(
            (unsigned int*)d_out, n);
    }
    head_mlp_kernel<<<32 * 8, 128, 0, stream>>>(
        nxyz2, feats2,
        Wp[6], scale[6], shift[6], Wp[7], scale[7], shift[7],
        Wp[8], scale[8], shift[8], (unsigned int*)d_out);
}